// NTMCell_56779467653269
// MI455X (gfx1250) — compile-verified
//
#include <hip/hip_runtime.h>
#include <hip/hip_bf16.h>
#include <math.h>

// Problem dims (from reference setup_inputs)
#define N 64
#define H 256
#define M 4096
#define D 128
#define EPSF 1e-8f

typedef __attribute__((ext_vector_type(2))) float v2f;
typedef __attribute__((ext_vector_type(8))) float v8f;

// d_out layout (floats): h_o[N*H] | C_new[N*M*D] | k[N*D] | r[N*D]
#define OUT_HO 0
#define OUT_C  (N*H)
#define OUT_K  (N*H + N*M*D)
#define OUT_R  (N*H + N*M*D + N*D)

// d_ws layout (floats)
#define WS_BETA  0
#define WS_NK    (N)
#define WS_W     (2*N)                 // slogit then w in place, N*M floats
#define WS_RPART (2*N + N*M)           // 8*N*D floats
#define WS_E     (2*N + N*M + 8*N*D)   // N*D
#define WS_V     (2*N + N*M + 9*N*D)   // N*D

static __device__ __forceinline__ v8f wmma_f32(v2f a, v2f b, v8f c) {
    return __builtin_amdgcn_wmma_f32_16x16x4_f32(false, a, false, b, (short)0, c, false, false);
}

// ---------------- K1: k = h_o_prev @ Wk.T + bk  (WMMA, 32 waves) ----------------
__global__ void k_gemm_kernel(const float* __restrict__ h,
                              const float* __restrict__ Wk,
                              const float* __restrict__ bk,
                              float* __restrict__ kout) {
    const int lane = threadIdx.x;            // 0..31, one wave per block
    const int nt = blockIdx.x >> 3;          // 0..3  (rows of 16 over N=64)
    const int dt = blockIdx.x & 7;           // 0..7  (cols of 16 over D=128)
    const int mrow = lane & 15;
    const int koff = (lane >> 4) * 2;
    v8f acc = {};
    for (int kk = 0; kk < H; kk += 4) {
        v2f a, b;
        a.x = h[(nt*16 + mrow)*H + kk + koff];
        a.y = h[(nt*16 + mrow)*H + kk + koff + 1];
        // B(K=h_local, N=d_local) = Wk[d, h]
        b.x = Wk[(dt*16 + mrow)*H + kk + koff];
        b.y = Wk[(dt*16 + mrow)*H + kk + koff + 1];
        acc = wmma_f32(a, b, acc);
    }
    const int col = dt*16 + mrow;
    const float bias = bk[col];
#pragma unroll
    for (int j = 0; j < 8; ++j) {
        int row = nt*16 + (lane >> 4)*8 + j;
        kout[row*D + col] = acc[j] + bias;
    }
}

// ---------------- K2: beta (=1+softplus) and ||k|| (tiny) ----------------
__global__ void beta_nk_kernel(const float* __restrict__ h,
                               const float* __restrict__ Wb,
                               const float* __restrict__ bb,
                               const float* __restrict__ k,
                               float* __restrict__ beta,
                               float* __restrict__ nk) {
    const int n = threadIdx.x;
    if (n >= N) return;
    float acc = bb[0];
    for (int j = 0; j < H; ++j) acc += h[n*H + j] * Wb[j];
    // reference beta == 1 + softplus(beta_pre)
    float sp = fmaxf(acc, 0.0f) + log1pf(expf(-fabsf(acc)));
    beta[n] = 1.0f + sp;
    float s = 0.0f;
    for (int d = 0; d < D; ++d) { float kv = k[n*D + d]; s += kv*kv; }
    nk[n] = sqrtf(s);
}

// ---------------- K3: scaled cosine logits (streams C once) ----------------
__global__ void dot_kernel(const float* __restrict__ C,
                           const float* __restrict__ k,
                           const float* __restrict__ beta,
                           const float* __restrict__ nk,
                           float* __restrict__ slogit) {
    const int tid = blockIdx.x * blockDim.x + threadIdx.x;
    const int row = tid >> 5;     // (n,m) row, one wave per row
    const int lane = tid & 31;
    const int n = row >> 12;      // row / M
    const float4 c4 = ((const float4*)C)[(long)row*(D/4) + lane];
    const float4 k4 = ((const float4*)k)[n*(D/4) + lane];
    float dot = c4.x*k4.x + c4.y*k4.y + c4.z*k4.z + c4.w*k4.w;
    float sq  = c4.x*c4.x + c4.y*c4.y + c4.z*c4.z + c4.w*c4.w;
#pragma unroll
    for (int off = 16; off >= 1; off >>= 1) {
        dot += __shfl_xor(dot, off, 32);
        sq  += __shfl_xor(sq,  off, 32);
    }
    if (lane == 0) {
        float nC = sqrtf(sq);
        float s = dot / (fmaxf(nC, EPSF) * fmaxf(nk[n], EPSF));
        slogit[row] = s * beta[n];
    }
}

// ---------------- K4: softmax over M per batch row (in place) ----------------
__global__ void softmax_kernel(float* __restrict__ w) {
    const int n = blockIdx.x;
    const int t = threadIdx.x;            // 256 threads, 16 elems each
    float* row = w + (long)n*M;
    __shared__ float sd[256];
    float x[16];
    float mx = -INFINITY;
#pragma unroll
    for (int i = 0; i < 16; ++i) { x[i] = row[t + i*256]; mx = fmaxf(mx, x[i]); }
    sd[t] = mx; __syncthreads();
    for (int s = 128; s > 0; s >>= 1) { if (t < s) sd[t] = fmaxf(sd[t], sd[t+s]); __syncthreads(); }
    mx = sd[0]; __syncthreads();
    float sum = 0.0f;
#pragma unroll
    for (int i = 0; i < 16; ++i) { x[i] = expf(x[i] - mx); sum += x[i]; }
    sd[t] = sum; __syncthreads();
    for (int s = 128; s > 0; s >>= 1) { if (t < s) sd[t] += sd[t+s]; __syncthreads(); }
    float inv = 1.0f / sd[0];
#pragma unroll
    for (int i = 0; i < 16; ++i) row[t + i*256] = x[i] * inv;
}

// ---------------- K5: r partials over 8 m-slices (deterministic) ----------------
__global__ void rpart_kernel(const float* __restrict__ C,
                             const float* __restrict__ w,
                             float* __restrict__ rpart) {
    const int n = blockIdx.x >> 3;
    const int slice = blockIdx.x & 7;
    const int d = threadIdx.x;            // 128 threads
    __shared__ float wl[512];
    const int m0 = slice * 512;
    for (int i = threadIdx.x; i < 512; i += 128) wl[i] = w[(long)n*M + m0 + i];
    __syncthreads();
    float acc = 0.0f;
    const float* Cr = C + ((long)(n*M + m0))*D + d;
    for (int m = 0; m < 512; ++m) acc += wl[m] * Cr[(long)m*D];
    rpart[(slice*N + n)*D + d] = acc;
}

__global__ void rreduce_kernel(const float* __restrict__ rpart,
                               float* __restrict__ r) {
    const int n = blockIdx.x;
    const int d = threadIdx.x;
    float acc = 0.0f;
#pragma unroll
    for (int s = 0; s < 8; ++s) acc += rpart[(s*N + n)*D + d];
    r[n*D + d] = acc;
}

// ---------------- K6: GRU cell (WMMA, 6 accumulators per wave) ----------------
__global__ void gru_kernel(const float* __restrict__ r,
                           const float* __restrict__ hprev,
                           const float* __restrict__ W_ih,
                           const float* __restrict__ W_hh,
                           const float* __restrict__ b_ih,
                           const float* __restrict__ b_hh,
                           float* __restrict__ h_o) {
    const int lane = threadIdx.x;
    const int nt = blockIdx.x >> 4;   // 0..3
    const int ct = blockIdx.x & 15;   // 0..15 (cols of 16 over H=256)
    const int mrow = lane & 15;
    const int koff = (lane >> 4) * 2;
    const int jr = ct*16 + mrow;      // B-matrix N index (gate column)
    v8f air = {}, aiz = {}, ain = {}, ahr = {}, ahz = {}, ahn = {};
    // gi = r @ W_ih.T : K over D
    for (int kk = 0; kk < D; kk += 4) {
        v2f a, br, bz, bn;
        a.x = r[(nt*16 + mrow)*D + kk + koff];
        a.y = r[(nt*16 + mrow)*D + kk + koff + 1];
        br.x = W_ih[(jr      )*D + kk + koff]; br.y = W_ih[(jr      )*D + kk + koff + 1];
        bz.x = W_ih[(jr + 256)*D + kk + koff]; bz.y = W_ih[(jr + 256)*D + kk + koff + 1];
        bn.x = W_ih[(jr + 512)*D + kk + koff]; bn.y = W_ih[(jr + 512)*D + kk + koff + 1];
        air = wmma_f32(a, br, air);
        aiz = wmma_f32(a, bz, aiz);
        ain = wmma_f32(a, bn, ain);
    }
    // gh = hprev @ W_hh.T : K over H
    for (int kk = 0; kk < H; kk += 4) {
        v2f a, br, bz, bn;
        a.x = hprev[(nt*16 + mrow)*H + kk + koff];
        a.y = hprev[(nt*16 + mrow)*H + kk + koff + 1];
        br.x = W_hh[(jr      )*H + kk + koff]; br.y = W_hh[(jr      )*H + kk + koff + 1];
        bz.x = W_hh[(jr + 256)*H + kk + koff]; bz.y = W_hh[(jr + 256)*H + kk + koff + 1];
        bn.x = W_hh[(jr + 512)*H + kk + koff]; bn.y = W_hh[(jr + 512)*H + kk + koff + 1];
        ahr = wmma_f32(a, br, ahr);
        ahz = wmma_f32(a, bz, ahz);
        ahn = wmma_f32(a, bn, ahn);
    }
    const int col = ct*16 + mrow;
#pragma unroll
    for (int j = 0; j < 8; ++j) {
        int row = nt*16 + (lane >> 4)*8 + j;
        float i_r = air[j] + b_ih[col];
        float i_z = aiz[j] + b_ih[256 + col];
        float i_n = ain[j] + b_ih[512 + col];
        float h_r = ahr[j] + b_hh[col];
        float h_z = ahz[j] + b_hh[256 + col];
        float h_n = ahn[j] + b_hh[512 + col];
        float rg = 1.0f / (1.0f + expf(-(i_r + h_r)));
        float z  = 1.0f / (1.0f + expf(-(i_z + h_z)));
        float ng = tanhf(i_n + rg * h_n);
        h_o[row*H + col] = (1.0f - z)*ng + z*hprev[row*H + col];
    }
}

// ---------------- K7: e = sigmoid(h_o@We.T+be), v = h_o@Wv.T+bv (WMMA) ----------------
__global__ void ev_kernel(const float* __restrict__ h_o,
                          const float* __restrict__ We,
                          const float* __restrict__ be,
                          const float* __restrict__ Wv,
                          const float* __restrict__ bv,
                          float* __restrict__ e,
                          float* __restrict__ v) {
    const int lane = threadIdx.x;
    const int nt = blockIdx.x >> 3;   // 0..3
    const int dt = blockIdx.x & 7;    // 0..7
    const int mrow = lane & 15;
    const int koff = (lane >> 4) * 2;
    const int jr = dt*16 + mrow;
    v8f ae = {}, av = {};
    for (int kk = 0; kk < H; kk += 4) {
        v2f a, b1, b2;
        a.x = h_o[(nt*16 + mrow)*H + kk + koff];
        a.y = h_o[(nt*16 + mrow)*H + kk + koff + 1];
        b1.x = We[jr*H + kk + koff]; b1.y = We[jr*H + kk + koff + 1];
        b2.x = Wv[jr*H + kk + koff]; b2.y = Wv[jr*H + kk + koff + 1];
        ae = wmma_f32(a, b1, ae);
        av = wmma_f32(a, b2, av);
    }
    const int col = dt*16 + mrow;
    const float be_c = be[col], bv_c = bv[col];
#pragma unroll
    for (int j = 0; j < 8; ++j) {
        int row = nt*16 + (lane >> 4)*8 + j;
        e[row*D + col] = 1.0f / (1.0f + expf(-(ae[j] + be_c)));
        v[row*D + col] = av[j] + bv_c;
    }
}

// ---------------- K8: C_new = C*(1 - w*e) + w*v (stream, float4) ----------------
__global__ void update_kernel(const float* __restrict__ C,
                              const float* __restrict__ w,
                              const float* __restrict__ e,
                              const float* __restrict__ v,
                              float* __restrict__ Cn) {
    const long idx = (long)blockIdx.x * blockDim.x + threadIdx.x;  // float4 index
    const int d4 = (int)(idx & (D/4 - 1));
    const long row = idx >> 5;                                      // (n,m)
    const int n = (int)(row >> 12);
    const float wv = w[row];
    const float4 c  = ((const float4*)C)[idx];
    const float4 ee = ((const float4*)e)[n*(D/4) + d4];
    const float4 vv = ((const float4*)v)[n*(D/4) + d4];
    float4 o;
    o.x = c.x * (1.0f - wv*ee.x) + wv*vv.x;
    o.y = c.y * (1.0f - wv*ee.y) + wv*vv.y;
    o.z = c.z * (1.0f - wv*ee.z) + wv*vv.z;
    o.w = c.w * (1.0f - wv*ee.w) + wv*vv.w;
    ((float4*)Cn)[idx] = o;
}

extern "C" void kernel_launch(void* const* d_in, const int* in_sizes, int n_in,
                              void* d_out, int out_size, void* d_ws, size_t ws_size,
                              hipStream_t stream) {
    const float* h_prev = (const float*)d_in[0];
    const float* C      = (const float*)d_in[1];
    const float* Wk     = (const float*)d_in[2];
    const float* bk     = (const float*)d_in[3];
    const float* Wb     = (const float*)d_in[4];
    const float* bb     = (const float*)d_in[5];
    const float* We     = (const float*)d_in[6];
    const float* be     = (const float*)d_in[7];
    const float* Wv     = (const float*)d_in[8];
    const float* bv     = (const float*)d_in[9];
    const float* W_ih   = (const float*)d_in[10];
    const float* W_hh   = (const float*)d_in[11];
    const float* b_ih   = (const float*)d_in[12];
    const float* b_hh   = (const float*)d_in[13];

    float* out = (float*)d_out;
    float* ho_out = out + OUT_HO;
    float* C_out  = out + OUT_C;
    float* k_out  = out + OUT_K;
    float* r_out  = out + OUT_R;

    float* ws    = (float*)d_ws;
    float* beta  = ws + WS_BETA;
    float* nk    = ws + WS_NK;
    float* w     = ws + WS_W;
    float* rpart = ws + WS_RPART;
    float* e_buf = ws + WS_E;
    float* v_buf = ws + WS_V;

    // 1) k = h @ Wk.T + bk  (WMMA)
    k_gemm_kernel<<<32, 32, 0, stream>>>(h_prev, Wk, bk, k_out);
    // 2) beta, ||k||
    beta_nk_kernel<<<1, 64, 0, stream>>>(h_prev, Wb, bb, k_out, beta, nk);
    // 3) logits = beta * cos-sim  (reads all of C)
    dot_kernel<<<(N*M)/8, 256, 0, stream>>>(C, k_out, beta, nk, w);
    // 4) softmax over M
    softmax_kernel<<<N, 256, 0, stream>>>(w);
    // 5) r = sum_m w*C  (reads all of C; L2-resident)
    rpart_kernel<<<N*8, 128, 0, stream>>>(C, w, rpart);
    rreduce_kernel<<<N, 128, 0, stream>>>(rpart, r_out);
    // 6) GRU (WMMA)
    gru_kernel<<<64, 32, 0, stream>>>(r_out, h_prev, W_ih, W_hh, b_ih, b_hh, ho_out);
    // 7) e, v (WMMA)
    ev_kernel<<<32, 32, 0, stream>>>(ho_out, We, be, Wv, bv, e_buf, v_buf);
    // 8) C_new  (reads all of C, writes all of C_new)
    update_kernel<<<(N*M*D/4)/256, 256, 0, stream>>>(C, w, e_buf, v_buf, C_out);
}